// ProposalLayer_18640158065287
// MI455X (gfx1250) — compile-verified
//
#include <hip/hip_runtime.h>
#include <math.h>

// ---------------- problem constants (from reference setup) ----------------
#define FEAT_STRIDE 16
#define A_ANCH      9
#define HH          64
#define WW          64
#define HW          (HH * WW)
#define K_PROP      (A_ANCH * HW)   // 36864 proposals per image
#define PADN        65536           // pow2 pad for bitonic sort
#define PRE_NMS     6000
#define POST_NMS    300
#define NMS_THR     0.7f
#define CHUNK       2048            // LDS-resident bitonic chunk
#define NEG_INF     (-3.402823466e+38f)
#define NT          375             // 6000/16 tiles per dimension
#define MROW        376             // u16 stride per mask row (pad to even)

typedef __attribute__((ext_vector_type(2))) float v2f;
typedef __attribute__((ext_vector_type(8))) float v8f;

// ---------------------------------------------------------------------------
// Kernel 1: fused anchor generation + bbox_transform_inv + clip + score gather
// ---------------------------------------------------------------------------
__global__ void proposals_kernel(const float* __restrict__ scores,
                                 const float* __restrict__ deltas,
                                 const float* __restrict__ img_info,
                                 const float* __restrict__ anchors,
                                 float* __restrict__ keys,
                                 int*   __restrict__ vals,
                                 float* __restrict__ boxes,
                                 int nB) {
  int gid = blockIdx.x * blockDim.x + threadIdx.x;
  if (gid >= nB * PADN) return;
  int b = gid >> 16;            // PADN == 65536
  int k = gid & (PADN - 1);
  if (k >= K_PROP) {            // padding for the pow2 sort
    keys[gid] = NEG_INF;
    vals[gid] = 0;
    return;
  }
  int s = k / A_ANCH;
  int a = k - s * A_ANCH;
  int h = s / WW;
  int w = s - h * WW;

  float shx = (float)(w * FEAT_STRIDE);
  float shy = (float)(h * FEAT_STRIDE);
  float ax1 = anchors[a * 4 + 0] + shx;
  float ay1 = anchors[a * 4 + 1] + shy;
  float ax2 = anchors[a * 4 + 2] + shx;
  float ay2 = anchors[a * 4 + 3] + shy;

  float wa = ax2 - ax1 + 1.0f;
  float ha = ay2 - ay1 + 1.0f;
  float cx = ax1 + 0.5f * wa;
  float cy = ay1 + 0.5f * ha;

  int dbase = ((b * (4 * A_ANCH)) + a * 4) * HW + s;
  float dx = deltas[dbase];
  float dy = deltas[dbase + HW];
  float dw = deltas[dbase + 2 * HW];
  float dh = deltas[dbase + 3 * HW];

  float pcx = dx * wa + cx;
  float pcy = dy * ha + cy;
  float pw  = expf(dw) * wa;
  float ph  = expf(dh) * ha;

  float x1 = pcx - 0.5f * pw;
  float y1 = pcy - 0.5f * ph;
  float x2 = pcx + 0.5f * pw;
  float y2 = pcy + 0.5f * ph;

  float maxx = img_info[1] - 1.0f;
  float maxy = img_info[0] - 1.0f;
  x1 = fminf(fmaxf(x1, 0.0f), maxx);
  y1 = fminf(fmaxf(y1, 0.0f), maxy);
  x2 = fminf(fmaxf(x2, 0.0f), maxx);
  y2 = fminf(fmaxf(y2, 0.0f), maxy);

  float4* bp = (float4*)(boxes + ((size_t)(b * K_PROP + k)) * 4);
  *bp = make_float4(x1, y1, x2, y2);

  keys[gid] = scores[((b * (2 * A_ANCH)) + A_ANCH + a) * HW + s];
  vals[gid] = k;
}

// ---------------------------------------------------------------------------
// Bitonic sort (descending), LDS-fused phases for j < CHUNK.
// ---------------------------------------------------------------------------
__device__ __forceinline__ void cmpxchg_lds(float* sk, int* sv, int li, int lj,
                                            bool descBlk) {
  float a = sk[li], b = sk[lj];
  bool sw = descBlk ? (a < b) : (a > b);
  if (sw) {
    sk[li] = b; sk[lj] = a;
    int t = sv[li]; sv[li] = sv[lj]; sv[lj] = t;
  }
}

__global__ void __launch_bounds__(1024)
bitonic_local_sort(float* __restrict__ keys, int* __restrict__ vals) {
  __shared__ float sk[CHUNK];
  __shared__ int   sv[CHUNK];
  const int nChunks = PADN / CHUNK;
  int b    = blockIdx.x / nChunks;
  int c    = blockIdx.x - b * nChunks;
  int base = c * CHUNK;
  size_t off = (size_t)b * PADN + base;
  int t = threadIdx.x;

  sk[t]        = keys[off + t];        sv[t]        = vals[off + t];
  sk[t + 1024] = keys[off + t + 1024]; sv[t + 1024] = vals[off + t + 1024];
  __syncthreads();

  for (int k = 2; k <= CHUNK; k <<= 1) {
    for (int j = k >> 1; j > 0; j >>= 1) {
      #pragma unroll
      for (int r = 0; r < 2; ++r) {
        int li = t + r * 1024;
        int lj = li ^ j;
        if (lj > li) cmpxchg_lds(sk, sv, li, lj, (((base + li) & k) == 0));
      }
      __syncthreads();
    }
  }
  keys[off + t]        = sk[t];        vals[off + t]        = sv[t];
  keys[off + t + 1024] = sk[t + 1024]; vals[off + t + 1024] = sv[t + 1024];
}

__global__ void bitonic_global_step(float* __restrict__ keys,
                                    int* __restrict__ vals,
                                    int k, int j, int nB) {
  int gid = blockIdx.x * blockDim.x + threadIdx.x;
  if (gid >= nB * PADN) return;
  int b = gid >> 16;
  int i = gid & (PADN - 1);
  int ix = i ^ j;
  if (ix > i) {
    size_t o = (size_t)b * PADN;
    float a = keys[o + i], c = keys[o + ix];
    bool descBlk = ((i & k) == 0);
    bool sw = descBlk ? (a < c) : (a > c);
    if (sw) {
      keys[o + i] = c; keys[o + ix] = a;
      int tv = vals[o + i]; vals[o + i] = vals[o + ix]; vals[o + ix] = tv;
    }
  }
}

__global__ void __launch_bounds__(1024)
bitonic_local_merge(float* __restrict__ keys, int* __restrict__ vals, int k) {
  __shared__ float sk[CHUNK];
  __shared__ int   sv[CHUNK];
  const int nChunks = PADN / CHUNK;
  int b    = blockIdx.x / nChunks;
  int c    = blockIdx.x - b * nChunks;
  int base = c * CHUNK;
  size_t off = (size_t)b * PADN + base;
  int t = threadIdx.x;

  sk[t]        = keys[off + t];        sv[t]        = vals[off + t];
  sk[t + 1024] = keys[off + t + 1024]; sv[t + 1024] = vals[off + t + 1024];
  __syncthreads();

  for (int j = CHUNK >> 1; j > 0; j >>= 1) {
    #pragma unroll
    for (int r = 0; r < 2; ++r) {
      int li = t + r * 1024;
      int lj = li ^ j;
      if (lj > li) cmpxchg_lds(sk, sv, li, lj, (((base + li) & k) == 0));
    }
    __syncthreads();
  }
  keys[off + t]        = sk[t];        vals[off + t]        = sv[t];
  keys[off + t + 1024] = sk[t + 1024]; vals[off + t + 1024] = sv[t + 1024];
}

// ---------------------------------------------------------------------------
// Device helper: stage the 6000 sorted boxes of image b into LDS (offset 0)
// via the gfx1250 async global->LDS path (per-lane gather, ASYNCcnt).
// Kernel must use ONLY dynamic LDS so raw offsets start at 0.
// ---------------------------------------------------------------------------
__device__ __forceinline__ void stage_boxes_async(const int* __restrict__ vals,
                                                  const float* __restrict__ boxes,
                                                  int b, int tid, int nthreads) {
  for (int i = tid; i < PRE_NMS; i += nthreads) {
    int v = vals[(size_t)b * PADN + i];
    unsigned long long ga =
        (unsigned long long)(const void*)(boxes + ((size_t)(b * K_PROP + v)) * 4);
    unsigned lds_addr = (unsigned)(i * 16);
    asm volatile("global_load_async_to_lds_b128 %0, %1, off"
                 :: "v"(lds_addr), "v"(ga) : "memory");
  }
  asm volatile("s_wait_asynccnt 0" ::: "memory");
}

__device__ __forceinline__ float box_area(const float* sb, int idx) {
  float x1 = sb[idx * 4 + 0], y1 = sb[idx * 4 + 1];
  float x2 = sb[idx * 4 + 2], y2 = sb[idx * 4 + 3];
  return (x2 - x1 + 1.0f) * (y2 - y1 + 1.0f);
}

// ---------------------------------------------------------------------------
// Kernel 3: all-pairs suppression bitmask, upper triangle, 16x16 tiles.
// One wave per tile. The union denominator area_i[m] + area_j[n] is a rank-2
// outer sum computed on the matrix unit: V_WMMA_F32_16X16X4_F32 with
//   A(16x4) = [area_i | 1 | 0 | 0],  B(4x16) = [1 ; area_j ; 0 ; 0]
// => D[m][n] = area_i[m] + area_j[n] (bit-exact f32 add). Intersection is
// min/max VALU. Per-row 16-bit masks packed with ballot, stored as u16.
// Block = 256 threads (8 waves) = one tile-row ti; boxes LDS-resident (96KB).
// ---------------------------------------------------------------------------
__global__ void __launch_bounds__(256)
nms_mask_kernel(const int* __restrict__ vals, const float* __restrict__ boxes,
                unsigned short* __restrict__ mask16, int nB) {
  extern __shared__ char smem[];
  float* sb = (float*)smem;                       // 6000*4 f32 @ LDS offset 0

  int b  = blockIdx.x / NT;
  int ti = blockIdx.x - b * NT;
  int tid  = threadIdx.x;
  int lane = tid & 31;
  int wv   = tid >> 5;
  bool hi  = lane >= 16;
  int l16  = lane & 15;

  stage_boxes_async(vals, boxes, b, tid, 256);
  __syncthreads();

  unsigned short* mrow = mask16 + (size_t)b * PRE_NMS * MROW;

  // A operand: lanes 0-15 -> {area_i[m], 1}; lanes 16-31 -> {0, 0} (K=2,3)
  v2f av;
  float area_i_lane = box_area(sb, ti * 16 + l16);
  av.x = hi ? 0.0f : area_i_lane;
  av.y = hi ? 0.0f : 1.0f;

  for (int tj = ti + wv; tj < NT; tj += 8) {
    // B operand: lanes 0-15 -> {1, area_j[n]}; lanes 16-31 -> {0, 0}
    float area_j_lane = box_area(sb, tj * 16 + l16);
    v2f bv;
    bv.x = hi ? 0.0f : 1.0f;
    bv.y = hi ? 0.0f : area_j_lane;

    v8f acc = {};
    acc = __builtin_amdgcn_wmma_f32_16x16x4_f32(
        /*neg_a=*/false, av, /*neg_b=*/false, bv,
        /*c_mod=*/(short)0, acc, /*reuse_a=*/false, /*reuse_b=*/false);

    // This lane's j-box (column n = lane&15) is fixed for the whole tile.
    int jdx = tj * 16 + l16;
    float bx1 = sb[jdx * 4 + 0], by1 = sb[jdx * 4 + 1];
    float bx2 = sb[jdx * 4 + 2], by2 = sb[jdx * 4 + 3];

    #pragma unroll
    for (int v = 0; v < 8; ++v) {
      int m   = v + (hi ? 8 : 0);          // D row for this lane/VGPR
      int idx = ti * 16 + m;
      float x1 = sb[idx * 4 + 0], y1 = sb[idx * 4 + 1];
      float x2 = sb[idx * 4 + 2], y2 = sb[idx * 4 + 3];
      float iw = fminf(x2, bx2) - fmaxf(x1, bx1) + 1.0f;
      float ih = fminf(y2, by2) - fmaxf(y1, by1) + 1.0f;
      float inter = fmaxf(0.0f, iw) * fmaxf(0.0f, ih);
      float uni = acc[v] - inter;          // area_i + area_j - inter (WMMA)
      bool pred = (inter / uni) > NMS_THR;
      unsigned bal = __builtin_amdgcn_ballot_w32(pred);
      if (lane == 0) {
        mrow[(size_t)(ti * 16 + v)     * MROW + tj] = (unsigned short)(bal & 0xFFFF);
        mrow[(size_t)(ti * 16 + v + 8) * MROW + tj] = (unsigned short)(bal >> 16);
      }
    }
  }
}

// ---------------------------------------------------------------------------
// Kernel 4: serial greedy reduce over the bitmask + output emit.
// One block per image. Suppression state = 188 words in LDS; only kept rows
// (<=6000, typically far fewer) are fetched from the L2-resident mask.
// Lower-triangle garbage is cleared by the exact j>i word mask.
// ---------------------------------------------------------------------------
#define SUPW 188   // ceil(6000/32)
__global__ void __launch_bounds__(256)
nms_reduce_kernel(const int* __restrict__ vals, const float* __restrict__ boxes,
                  const unsigned short* __restrict__ mask16,
                  float* __restrict__ out) {
  extern __shared__ char smem[];
  float*    sb   = (float*)smem;                      // 6000*4 f32 @ 0
  unsigned* supw = (unsigned*)(smem + PRE_NMS * 16);  // 188 words
  int*      kcnt = (int*)(smem + PRE_NMS * 16 + SUPW * 4);

  int b   = blockIdx.x;
  int tid = threadIdx.x;

  stage_boxes_async(vals, boxes, b, tid, 256);
  for (int w = tid; w < SUPW; w += 256) supw[w] = 0;
  if (tid == 0) *kcnt = 0;
  __syncthreads();

  const unsigned short* mrow = mask16 + (size_t)b * PRE_NMS * MROW;

  for (int i = 0; i < PRE_NMS; ++i) {
    bool supp = (supw[i >> 5] >> (i & 31)) & 1u;   // uniform LDS read
    if (!supp) {
      if (tid == 0) {
        int c = *kcnt;
        if (c < POST_NMS) {
          float* o = out + ((size_t)(b * POST_NMS + c)) * 5;
          o[0] = (float)b;
          o[1] = sb[i * 4 + 0]; o[2] = sb[i * 4 + 1];
          o[3] = sb[i * 4 + 2]; o[4] = sb[i * 4 + 3];
        }
        *kcnt = c + 1;
      }
      // suppressed |= over[i] & (j > i)
      const unsigned* row = (const unsigned*)(mrow + (size_t)i * MROW);
      for (int w = tid; w < SUPW; w += 256) {
        int base = w << 5;
        if (base + 31 > i) {
          unsigned m = row[w];
          if (base <= i) m &= (0xFFFFFFFEu << (i - base));  // keep bits j > i
          supw[w] |= m;   // each word owned by one thread; bit i untouched
        }
      }
    }
    __syncthreads();
  }

  int c = *kcnt;
  if (c > POST_NMS) c = POST_NMS;
  for (int r = c + tid; r < POST_NMS; r += 256) {
    float* o = out + ((size_t)(b * POST_NMS + r)) * 5;
    o[0] = (float)b; o[1] = 0.0f; o[2] = 0.0f; o[3] = 0.0f; o[4] = 0.0f;
  }
}

// ---------------------------------------------------------------------------
extern "C" void kernel_launch(void* const* d_in, const int* in_sizes, int n_in,
                              void* d_out, int out_size, void* d_ws, size_t ws_size,
                              hipStream_t stream) {
  const float* scores   = (const float*)d_in[0];  // (B,18,64,64)
  const float* deltas   = (const float*)d_in[1];  // (B,36,64,64)
  const float* img_info = (const float*)d_in[2];  // (2,)
  const float* anchors  = (const float*)d_in[3];  // (9,4)
  float* out = (float*)d_out;                     // (B,300,5)

  int nB = in_sizes[0] / (2 * A_ANCH * HW);

  // Workspace layout (~22.5 MB for B=4):
  float* keys  = (float*)d_ws;                              // B*PADN f32
  int*   vals  = (int*)(keys + (size_t)nB * PADN);          // B*PADN i32
  float* boxes = (float*)(vals + (size_t)nB * PADN);        // B*K_PROP*4 f32
  unsigned short* mask16 =
      (unsigned short*)(boxes + (size_t)nB * K_PROP * 4);   // B*6000*376 u16

  int total = nB * PADN;
  proposals_kernel<<<(total + 255) / 256, 256, 0, stream>>>(
      scores, deltas, img_info, anchors, keys, vals, boxes, nB);

  bitonic_local_sort<<<nB * (PADN / CHUNK), 1024, 0, stream>>>(keys, vals);
  for (int k = 2 * CHUNK; k <= PADN; k <<= 1) {
    for (int j = k >> 1; j >= CHUNK; j >>= 1)
      bitonic_global_step<<<(total + 255) / 256, 256, 0, stream>>>(keys, vals,
                                                                   k, j, nB);
    bitonic_local_merge<<<nB * (PADN / CHUNK), 1024, 0, stream>>>(keys, vals, k);
  }

  // WMMA-tiled suppression bitmask: one block per (image, tile-row).
  size_t shmask = (size_t)PRE_NMS * 16;                       // 96KB boxes
  nms_mask_kernel<<<nB * NT, 256, shmask, stream>>>(vals, boxes, mask16, nB);

  // Serial bit-OR reduce + emit.
  size_t shred = (size_t)PRE_NMS * 16 + SUPW * 4 + 16;
  nms_reduce_kernel<<<nB, 256, shred, stream>>>(vals, boxes, mask16, out);
}